// VectorQuantizer_12378095747428
// MI455X (gfx1250) — compile-verified
//
#include <hip/hip_runtime.h>

typedef float v2f __attribute__((ext_vector_type(2)));
typedef float v8f __attribute__((ext_vector_type(8)));

#define BATCH   32
#define D_DIM   64
#define HW      1024
#define N_ROWS  (BATCH * HW)       // 32768
#define K_EMB   1024
#define ELEMS   (N_ROWS * D_DIM)   // 2097152

// d_out float offsets (tuple order: distance, quantize_out, loss, encodings, indices, perplexity)
#define OFF_DIST  0
#define OFF_QUANT 33554432ll
#define OFF_LOSS  35651584ll
#define OFF_ENC   35651585ll
#define OFF_IDX   69206017ll
#define OFF_PERP  69238785ll

// workspace byte offsets
#define WS_ZNORM 0
#define WS_ENORM 131072
#define WS_IDX   135168
#define WS_CNT   266240
#define WS_ACC   270336

__global__ __launch_bounds__(1024)
void vq_zero_kernel(float* __restrict__ counts, float* __restrict__ acc) {
    int t = threadIdx.x;
    if (t < K_EMB) counts[t] = 0.0f;
    if (t == 0) acc[0] = 0.0f;
}

// per-row ||z||^2 and per-code ||e||^2
__global__ __launch_bounds__(256)
void vq_norms_kernel(const float* __restrict__ x, const float* __restrict__ emb,
                     float* __restrict__ znorm, float* __restrict__ enorm) {
    int gid = blockIdx.x * 256 + threadIdx.x;
    if (gid < N_ROWS) {
        int b = gid >> 10, hw = gid & 1023;
        const float* px = x + (size_t)b * (D_DIM * HW) + hw;
        float s = 0.0f;
        #pragma unroll 8
        for (int d = 0; d < D_DIM; ++d) { float v = px[(size_t)d * HW]; s = fmaf(v, v, s); }
        znorm[gid] = s;
    } else if (gid < N_ROWS + K_EMB) {
        int k = gid - N_ROWS;
        const float* pe = emb + k * D_DIM;
        float s = 0.0f;
        #pragma unroll 8
        for (int d = 0; d < D_DIM; ++d) s = fmaf(pe[d], pe[d], s);
        enorm[k] = s;
    }
}

// Main: distance via V_WMMA_F32_16X16X4_F32, argmin, encodings, indices, histogram.
// Block = 256 threads = 8 wave32s; block owns rows [blockIdx.x*16, +16); wave w owns cols [w*128, +128).
__global__ __launch_bounds__(256)
void vq_gemm_kernel(const float* __restrict__ x, const float* __restrict__ emb,
                    const float* __restrict__ znorm, const float* __restrict__ enorm,
                    float* __restrict__ dist, float* __restrict__ enc,
                    float* __restrict__ idxf, int* __restrict__ idxw,
                    float* __restrict__ counts) {
    __shared__ float sval[8][16];
    __shared__ int   scol[8][16];
    __shared__ int   sidx[16];

    const int tid  = threadIdx.x;
    const int lane = tid & 31;
    const int w    = tid >> 5;
    const int half = lane >> 4;        // 0: lanes 0-15, 1: lanes 16-31
    const int nlo  = lane & 15;
    const int rowBase = blockIdx.x * 16;
    const int colBase = w * 128;

    v8f acc[8];
    #pragma unroll
    for (int t = 0; t < 8; ++t)
        #pragma unroll
        for (int v = 0; v < 8; ++v) acc[t][v] = 0.0f;

    // A fragment addressing: 16x4 f32, lane holds M=nlo; VGPR j holds K = 4*kk + 2*half + j
    const int aRow = rowBase + nlo;
    const int ab   = aRow >> 10;
    const int ahw  = aRow & 1023;
    const float* ax = x + (size_t)ab * (D_DIM * HW) + ahw;
    const int koff = half * 2;

    #pragma unroll
    for (int kk = 0; kk < 16; ++kk) {
        const int kb = kk * 4 + koff;
        v2f a;
        a.x = ax[(size_t)kb * HW];
        a.y = ax[(size_t)(kb + 1) * HW];
        #pragma unroll
        for (int t = 0; t < 8; ++t) {
            const int col = colBase + t * 16 + nlo;     // B: 4x16, lane holds N=nlo
            v2f bf = *(const v2f*)(emb + col * D_DIM + kb);
            acc[t] = __builtin_amdgcn_wmma_f32_16x16x4_f32(
                false, a, false, bf, (short)0, acc[t], false, false);
        }
    }

    // C/D layout: lane half h, VGPR v -> row M = v + 8*h; N = nlo (+tile offset)
    float zn[8];
    #pragma unroll
    for (int v = 0; v < 8; ++v) zn[v] = znorm[rowBase + (half << 3) + v];

    float mval[8]; int mcol[8];
    #pragma unroll
    for (int v = 0; v < 8; ++v) { mval[v] = 3.402823e38f; mcol[v] = 0; }

    #pragma unroll
    for (int t = 0; t < 8; ++t) {
        const int col = colBase + t * 16 + nlo;
        const float en = enorm[col];
        #pragma unroll
        for (int v = 0; v < 8; ++v) {
            const float val = fmaf(-2.0f, acc[t][v], zn[v] + en);
            const size_t row = rowBase + (half << 3) + v;
            __builtin_nontemporal_store(val, dist + row * K_EMB + col);
            if (val < mval[v] || (val == mval[v] && col < mcol[v])) { mval[v] = val; mcol[v] = col; }
        }
    }

    // reduce across the 16 lanes that share each row (xor masks stay within each half)
    #pragma unroll
    for (int off = 1; off < 16; off <<= 1) {
        #pragma unroll
        for (int v = 0; v < 8; ++v) {
            float ov = __shfl_xor(mval[v], off, 32);
            int   oc = __shfl_xor(mcol[v], off, 32);
            if (ov < mval[v] || (ov == mval[v] && oc < mcol[v])) { mval[v] = ov; mcol[v] = oc; }
        }
    }
    if (nlo == 0) {
        #pragma unroll
        for (int v = 0; v < 8; ++v) {
            sval[w][(half << 3) + v] = mval[v];
            scol[w][(half << 3) + v] = mcol[v];
        }
    }
    __syncthreads();

    if (tid < 16) {
        float best = sval[0][tid]; int bc = scol[0][tid];
        #pragma unroll
        for (int ww = 1; ww < 8; ++ww) {
            float ov = sval[ww][tid]; int oc = scol[ww][tid];
            if (ov < best || (ov == best && oc < bc)) { best = ov; bc = oc; }
        }
        sidx[tid] = bc;
        const int row = rowBase + tid;
        idxw[row] = bc;
        idxf[row] = (float)bc;
        atomicAdd(&counts[bc], 1.0f);
    }
    __syncthreads();

    // one-hot encodings (base offset is odd -> scalar dword stores, coalesced along cols)
    #pragma unroll 4
    for (int j = tid; j < 16 * K_EMB; j += 256) {
        const int r = j >> 10, c = j & 1023;
        const float v = (c == sidx[r]) ? 1.0f : 0.0f;
        __builtin_nontemporal_store(v, enc + (size_t)(rowBase + r) * K_EMB + c);
    }
}

// gather-quantize to NCHW + squared-error sum
__global__ __launch_bounds__(256)
void vq_quant_kernel(const float* __restrict__ x, const float* __restrict__ emb,
                     const int* __restrict__ idxw, float* __restrict__ outq,
                     float* __restrict__ acc) {
    __shared__ float red[8];
    const int e = blockIdx.x * 256 + threadIdx.x;     // NCHW flat index
    const int b  = e >> 16;
    const int d  = (e >> 10) & 63;
    const int hw = e & 1023;
    const int n  = (b << 10) | hw;
    const int k  = idxw[n];
    const float q  = emb[k * D_DIM + d];
    const float xv = x[e];
    __builtin_nontemporal_store(q, outq + e);
    const float diff = q - xv;
    float s = diff * diff;
    #pragma unroll
    for (int off = 16; off >= 1; off >>= 1) s += __shfl_down(s, off, 32);
    if ((threadIdx.x & 31) == 0) red[threadIdx.x >> 5] = s;
    __syncthreads();
    if (threadIdx.x == 0) {
        float t = 0.0f;
        #pragma unroll
        for (int i = 0; i < 8; ++i) t += red[i];
        atomicAdd(acc, t);
    }
}

__global__ __launch_bounds__(1024)
void vq_final_kernel(const float* __restrict__ counts, const float* __restrict__ acc,
                     float* __restrict__ loss_out, float* __restrict__ perp_out) {
    __shared__ float red[32];
    const int t = threadIdx.x;
    const float p = counts[t] * (1.0f / (float)N_ROWS);
    float term = p * __logf(p + 1e-10f);
    #pragma unroll
    for (int off = 16; off >= 1; off >>= 1) term += __shfl_down(term, off, 32);
    if ((t & 31) == 0) red[t >> 5] = term;
    __syncthreads();
    if (t < 32) {
        float s = red[t];
        #pragma unroll
        for (int off = 16; off >= 1; off >>= 1) s += __shfl_down(s, off, 32);
        if (t == 0) {
            perp_out[0] = __expf(-s);
            // q_latent_loss + COMMITMENT_COST * e_latent_loss = 1.25 * MSE
            loss_out[0] = 1.25f * acc[0] * (1.0f / (float)ELEMS);
        }
    }
}

extern "C" void kernel_launch(void* const* d_in, const int* in_sizes, int n_in,
                              void* d_out, int out_size, void* d_ws, size_t ws_size,
                              hipStream_t stream) {
    const float* x   = (const float*)d_in[0];   // [32,64,32,32]
    const float* emb = (const float*)d_in[1];   // [1024,64]
    float* out = (float*)d_out;
    char*  ws  = (char*)d_ws;

    float* znorm  = (float*)(ws + WS_ZNORM);
    float* enorm  = (float*)(ws + WS_ENORM);
    int*   idxw   = (int*)  (ws + WS_IDX);
    float* counts = (float*)(ws + WS_CNT);
    float* acc    = (float*)(ws + WS_ACC);

    vq_zero_kernel<<<1, 1024, 0, stream>>>(counts, acc);
    vq_norms_kernel<<<(N_ROWS + K_EMB + 255) / 256, 256, 0, stream>>>(x, emb, znorm, enorm);
    vq_gemm_kernel<<<N_ROWS / 16, 256, 0, stream>>>(
        x, emb, znorm, enorm,
        out + OFF_DIST, out + OFF_ENC, out + OFF_IDX, idxw, counts);
    vq_quant_kernel<<<ELEMS / 256, 256, 0, stream>>>(x, emb, idxw, out + OFF_QUANT, acc);
    vq_final_kernel<<<1, 1024, 0, stream>>>(counts, acc, out + OFF_LOSS, out + OFF_PERP);
}